// DistMultPredictor_28278064677215
// MI455X (gfx1250) — compile-verified
//
#include <hip/hip_runtime.h>
#include <hip/hip_bf16.h>

// DistMult edge scoring, MI455X (gfx1250, wave32).
//
// score[r,e] = sigmoid( sum_d h[src[r,e],d] * W[r,d] * h[dst[r,e],d] ),  D=128
//
// Strategy: memory (L2-gather) bound -> per-lane b64 row-contiguous loads with
// a fully unrolled K loop for MLP; the 16-edge batched dot-product reduction is
// done on the matrix unit as diag( (Hu o W) x Hv^T ) via V_WMMA_F32_16X16X4_F32
// (full f32 precision; the 16x compute waste is free at this arithmetic
// intensity). W rows are staged in LDS (3 KB) and read with ds_load_b64.

#define N_REL     6
#define N_HID     128
#define EDGE_TILE 16

typedef __attribute__((ext_vector_type(2))) float v2f;
typedef __attribute__((ext_vector_type(8))) float v8f;

__global__ void __launch_bounds__(256)
distmult_wmma_kernel(const float* __restrict__ h,
                     const float* __restrict__ W,
                     const long long* __restrict__ src,
                     const long long* __restrict__ dst,
                     float* __restrict__ out,
                     int E, int nTiles, int tilesPerRel)
{
    // Stage all relation weight rows in LDS once per block (6*128*4B = 3 KB).
    __shared__ float ldsW[N_REL * N_HID];
    for (int i = threadIdx.x; i < N_REL * N_HID; i += blockDim.x)
        ldsW[i] = W[i];
    __syncthreads();

    const int lane = threadIdx.x & 31;
    const int kh   = lane >> 4;   // 0 -> K pair {4k,4k+1}, 1 -> K pair {4k+2,4k+3}
    const int m    = lane & 15;   // edge-in-tile (A row index == B column index)

    const int wave = (int)((blockIdx.x * blockDim.x + threadIdx.x) >> 5);
    if (wave >= nTiles) return;   // wave-uniform: EXEC stays all-1s for WMMA

    const int r     = wave / tilesPerRel;
    const int ebase = (wave - r * tilesPerRel) * EDGE_TILE;
    const int em    = min(ebase + m, E - 1);          // clamp (E%16==0 -> no-op)
    const long long eidx = (long long)r * E + em;

    const long long s = src[eidx];
    const long long d = dst[eidx];

    // Lane L and lane L+16 read the SAME row (edge m) at adjacent 8B offsets:
    // the pair covers 16 contiguous bytes per k-step; the unrolled loop streams
    // the whole 512B row. This matches both the 16x4 A and 4x16 B f32 WMMA
    // VGPR layouts, so no cross-lane shuffling is needed on the load path.
    const float* __restrict__ hu = h + (size_t)s * N_HID + kh * 2;
    const float* __restrict__ hv = h + (size_t)d * N_HID + kh * 2;
    const float* __restrict__ wr = ldsW + r * N_HID + kh * 2;

    v8f c = {};  // 16x16 f32 accumulator (diagonal = the 16 edge scores)

    #pragma unroll
    for (int kk = 0; kk < N_HID / 4; ++kk) {          // 32 x K=4 WMMA steps
        v2f a = *(const v2f*)(hu + kk * 4);           // global_load_b64
        v2f b = *(const v2f*)(hv + kk * 4);           // global_load_b64
        v2f w = *(const v2f*)(wr + kk * 4);           // ds_load_b64
        a *= w;                                       // fold W into A operand
        // D = A(16x4) x B(4x16) + C, full f32 (RNE) on the matrix unit.
        c = __builtin_amdgcn_wmma_f32_16x16x4_f32(
                /*neg_a=*/false, a, /*neg_b=*/false, b,
                /*c_mod=*/(short)0, c, /*reuse_a=*/false, /*reuse_b=*/false);
    }

    // Extract diag(C): C VGPR v holds (M=v, N=lane) for lanes 0-15 and
    // (M=8+v, N=lane-16) for lanes 16-31 -> diag(v) @ lane v, diag(8+v) @ lane 24+v.
    float sc = 0.0f;
    #pragma unroll
    for (int v = 0; v < 8; ++v) {
        float d0 = __shfl(c[v], v, 32);
        float d1 = __shfl(c[v], 24 + v, 32);
        sc = (lane == v)     ? d0 : sc;
        sc = (lane == 8 + v) ? d1 : sc;
    }

    if (lane < 16 && (ebase + lane) < E) {
        float sig = 1.0f / (1.0f + __expf(-sc));
        out[(long long)r * E + ebase + lane] = sig;
    }
}

extern "C" void kernel_launch(void* const* d_in, const int* in_sizes, int n_in,
                              void* d_out, int out_size, void* d_ws, size_t ws_size,
                              hipStream_t stream)
{
    const float*     h   = (const float*)d_in[0];      // [100000,128] f32
    const float*     W   = (const float*)d_in[1];      // [6,128] f32
    const long long* src = (const long long*)d_in[2];  // [6,E] int64
    const long long* dst = (const long long*)d_in[3];  // [6,E] int64
    float*           out = (float*)d_out;              // [6,E] f32

    const int E           = in_sizes[2] / N_REL;                     // 200000
    const int tilesPerRel = (E + EDGE_TILE - 1) / EDGE_TILE;         // 12500
    const int nTiles      = N_REL * tilesPerRel;                     // 75000
    const int wavesPerBlk = 256 / 32;                                // 8
    const int blocks      = (nTiles + wavesPerBlk - 1) / wavesPerBlk; // 9375

    distmult_wmma_kernel<<<blocks, 256, 0, stream>>>(h, W, src, dst, out,
                                                     E, nTiles, tilesPerRel);
}